// KNN_1675037245629
// MI455X (gfx1250) — compile-verified
//
#include <hip/hip_runtime.h>
#include <hip/hip_bf16.h>
#include <math.h>

// ---------------------------------------------------------------------------
// KNN label aggregation for MI455X (gfx1250, wave32, WMMA).
// Pipeline:
//   1) pad projector [3072x30] -> projP [3072x32] (zeros in cols 30,31), psum = colsum(projP)
//   2) k_rowpass(raw): ONE pass over 614MB computing per-row sum, sumsq and raw@projP
//      via v_wmma_f32_16x16x4_f32 (16 rows per wave, K=3072 in steps of 4).
//   3) deterministic single-block reduction -> global mean m
//   4) fixup: data[n] = (rawproj[n] - m*psum)/||raw_n - m||, dd[n]=||data_n||^2
//      (rows >= N padded with data=0, dd=1e30 so they never win top-k)
//   5) k_rowpass(x) + per-row fixup -> px [2048x32], qq[b]=||px_b||^2
//   6) k_knn: 128 WGs x 512 thr; WG owns 16 queries. Per 512-candidate chunk:
//      16 waves x 2 WMMA tiles (8 chained 16x16x4 each) -> d2 into LDS,
//      then wave-per-query scan with register sorted top-15; final shfl-based
//      32-way merge + exp/log label aggregation.
// WMMA f32 16x16x4 layouts (ISA 7.12.2): A/B: K = vgpr + 2*(lane>=16),
// M or N = lane%16.  C/D: vgpr r -> (M = r + 8*(lane>=16), N = lane%16).
// ---------------------------------------------------------------------------

typedef __attribute__((ext_vector_type(2))) float v2f;
typedef __attribute__((ext_vector_type(8))) float v8f;

#define D_DIM  3072
#define P_PAD  32
#define N_REF  50000
#define B_Q    2048
#define C_LAB  10
#define K_TOP  15
#define CHUNK  512
#define N_PAD  (98 * CHUNK)   /* 50176 = 98 chunks of 512 */
#define LDS_STRIDE 516        /* 512 + 4 pad: conflict-free (516 % 64 == 4) */

static __device__ __forceinline__ v8f wmma4(v2f a, v2f b, v8f c) {
  // D = A(16x4) * B(4x16) + C(16x16), all f32
  return __builtin_amdgcn_wmma_f32_16x16x4_f32(false, a, false, b, (short)0, c,
                                               false, false);
}

// ---- 1) projector padding + column sums -----------------------------------
__global__ void k_pad_projector(const float* __restrict__ proj,
                                float* __restrict__ projP) {
  int i = blockIdx.x * 256 + threadIdx.x;
  if (i >= D_DIM * P_PAD) return;
  int r = i >> 5, c = i & 31;
  projP[i] = (c < 30) ? proj[r * 30 + c] : 0.f;
}

__global__ void k_psum(const float* __restrict__ projP,
                       float* __restrict__ psum) {
  int c = threadIdx.x;  // 32 threads
  float s = 0.f;
  for (int r = 0; r < D_DIM; ++r) s += projP[r * P_PAD + c];
  psum[c] = s;
}

// ---- 2) fused row-stats + projection GEMM (one wave = 16 rows) ------------
__global__ void k_rowpass(const float* __restrict__ src,
                          const float* __restrict__ projP,
                          float* __restrict__ rowproj,
                          float* __restrict__ rowsum,
                          float* __restrict__ rowsumsq) {
  const int lane = threadIdx.x & 31;
  const int lo = lane & 15, hi = lane >> 4;
  const int rb = blockIdx.x * 16;
  const float* arow = src + (size_t)(rb + lo) * D_DIM + 2 * hi;

  v8f C0 = {};  // P columns 0..15
  v8f C1 = {};  // P columns 16..31
  float s = 0.f, q = 0.f;

  for (int d4 = 0; d4 < D_DIM / 4; ++d4) {
    v2f A = *(const v2f*)(arow + 4 * d4);          // raw[row][4d4+2hi .. +1]
    s += A.x + A.y;
    q = fmaf(A.x, A.x, q);
    q = fmaf(A.y, A.y, q);
    const float* bp = projP + (size_t)(4 * d4 + 2 * hi) * P_PAD + lo;
    v2f B0; B0.x = bp[0];  B0.y = bp[P_PAD];
    v2f B1; B1.x = bp[16]; B1.y = bp[16 + P_PAD];
    C0 = wmma4(A, B0, C0);
    C1 = wmma4(A, B1, C1);
  }
  // lanes l and l+16 together cover a full row (d%4 in {0,1} vs {2,3})
  s += __shfl_xor(s, 16);
  q += __shfl_xor(q, 16);
  if (lane < 16) { rowsum[rb + lane] = s; rowsumsq[rb + lane] = q; }

#pragma unroll
  for (int r = 0; r < 8; ++r) {
    rowproj[(size_t)(rb + r + 8 * hi) * P_PAD + lo]      = C0[r];
    rowproj[(size_t)(rb + r + 8 * hi) * P_PAD + 16 + lo] = C1[r];
  }
}

// ---- 3) deterministic global mean -----------------------------------------
__global__ void k_mean(const float* __restrict__ rowsum,
                       float* __restrict__ meanout) {
  __shared__ float red[1024];
  int tid = threadIdx.x;
  float t = 0.f;
  for (int i = tid; i < N_REF; i += 1024) t += rowsum[i];
  red[tid] = t;
  __syncthreads();
  for (int s = 512; s > 0; s >>= 1) {
    if (tid < s) red[tid] += red[tid + s];
    __syncthreads();
  }
  if (tid == 0) meanout[0] = red[0] / ((float)N_REF * (float)D_DIM);
}

// ---- 4) bank fixup: normalize projections, pad tail ------------------------
__global__ void k_fixup_data(const float* __restrict__ rowproj,
                             const float* __restrict__ rowsum,
                             const float* __restrict__ rowsumsq,
                             const float* __restrict__ psum,
                             const float* __restrict__ meanp,
                             float* __restrict__ data,
                             float* __restrict__ dd) {
  int n = blockIdx.x * 256 + threadIdx.x;
  if (n >= N_PAD) return;
  if (n < N_REF) {
    float m = meanp[0];
    float sv = rowsum[n], qv = rowsumsq[n];
    float nrm2 = qv - 2.f * m * sv + (float)D_DIM * m * m;  // ||raw_n - m||^2
    float inv = 1.f / sqrtf(fmaxf(nrm2, 1e-24f));
    float acc = 0.f;
#pragma unroll
    for (int p = 0; p < P_PAD; ++p) {
      float v = (rowproj[(size_t)n * P_PAD + p] - m * psum[p]) * inv;
      data[(size_t)n * P_PAD + p] = v;
      acc = fmaf(v, v, acc);
    }
    dd[n] = acc;
  } else {
#pragma unroll
    for (int p = 0; p < P_PAD; ++p) data[(size_t)n * P_PAD + p] = 0.f;
    dd[n] = 1e30f;  // padded candidates never selected
  }
}

// ---- 5) query fixup (per-row mean) ----------------------------------------
__global__ void k_fixup_x(const float* __restrict__ xproj,
                          const float* __restrict__ xsum,
                          const float* __restrict__ xsumsq,
                          const float* __restrict__ psum,
                          float* __restrict__ px, float* __restrict__ qq) {
  int b = blockIdx.x * 256 + threadIdx.x;
  if (b >= B_Q) return;
  float sv = xsum[b], qv2 = xsumsq[b];
  float mean = sv / (float)D_DIM;
  float nrm2 = qv2 - sv * sv / (float)D_DIM;  // ||x_b - mean||^2
  float inv = 1.f / sqrtf(fmaxf(nrm2, 1e-24f));
  float acc = 0.f;
#pragma unroll
  for (int p = 0; p < P_PAD; ++p) {
    float v = (xproj[(size_t)b * P_PAD + p] - mean * psum[p]) * inv;
    px[(size_t)b * P_PAD + p] = v;
    acc = fmaf(v, v, acc);
  }
  qq[b] = acc;
}

// ---- 6) fused WMMA cdist + top-15 + label aggregation ----------------------
__global__ void __launch_bounds__(512, 1) k_knn(
    const float* __restrict__ px, const float* __restrict__ qq,
    const float* __restrict__ data, const float* __restrict__ dd,
    const float* __restrict__ labels, float* __restrict__ out) {
  // 61,440B static LDS: phase uses 16*516 floats; merge reuses 16*32*30 floats
  __shared__ float smem[16 * 32 * 30];

  const int tid = threadIdx.x;
  const int wave = tid >> 5, lane = tid & 31;
  const int lo = lane & 15, hi = lane >> 4;
  const int qbase = blockIdx.x * 16;

  // Preload this block's A matrix (16 queries x 32 padded dims) once.
  v2f A[8];
#pragma unroll
  for (int s = 0; s < 8; ++s)
    A[s] = *(const v2f*)(px + (size_t)(qbase + lo) * P_PAD + 4 * s + 2 * hi);
  float qv[8];
#pragma unroll
  for (int r = 0; r < 8; ++r) qv[r] = qq[qbase + r + 8 * hi];

  float bd[K_TOP];
  int bi[K_TOP];
#pragma unroll
  for (int j = 0; j < K_TOP; ++j) { bd[j] = 1e30f; bi[j] = 0; }

  for (int chunk = 0; chunk < N_PAD / CHUNK; ++chunk) {
    const int cb = chunk * CHUNK;
    // Phase A: each wave computes 2 column tiles (32 candidates) of d2.
#pragma unroll
    for (int t = 0; t < 2; ++t) {
      const int nb = cb + wave * 32 + t * 16;
      const float* brow = data + (size_t)(nb + lo) * P_PAD + 2 * hi;
      v8f C = {};
#pragma unroll
      for (int s = 0; s < 8; ++s) {
        v2f B = *(const v2f*)(brow + 4 * s);  // data[n][4s+2hi .. +1]
        C = wmma4(A[s], B, C);                // S += px . data
      }
      float ddv = dd[nb + lo];
      const int col = wave * 32 + t * 16 + lo;
#pragma unroll
      for (int r = 0; r < 8; ++r) {
        float d2 = qv[r] + ddv - 2.f * C[r];
        smem[(r + 8 * hi) * LDS_STRIDE + col] = d2;
      }
    }
    __syncthreads();
    // Phase B: wave w scans query row w, running sorted top-15 per lane.
    const float* rowp = smem + wave * LDS_STRIDE;
#pragma unroll
    for (int i = 0; i < CHUNK / 32; ++i) {
      const int col = lane + 32 * i;
      float d2 = rowp[col];
      if (d2 < bd[K_TOP - 1]) {
        float cd = d2;
        int ci = cb + col;
#pragma unroll
        for (int j = 0; j < K_TOP; ++j) {
          if (cd < bd[j]) {
            float tf = bd[j]; bd[j] = cd; cd = tf;
            int ti = bi[j];  bi[j] = ci; ci = ti;
          }
        }
      }
    }
    __syncthreads();
  }

  // Merge 32 sorted lane-lists per wave: dump to LDS, 15x shfl arg-min.
  const int mbase = wave * 960 + lane * 30;
#pragma unroll
  for (int j = 0; j < K_TOP; ++j) {
    smem[mbase + 2 * j]     = bd[j];
    smem[mbase + 2 * j + 1] = __int_as_float(bi[j]);
  }
  __syncthreads();

  int hp = 0;
  float acc = 0.f;
  for (int sel = 0; sel < K_TOP; ++sel) {
    float v = (hp < K_TOP) ? smem[mbase + 2 * hp] : 3.0e38f;
    int src = lane;
#pragma unroll
    for (int off = 16; off >= 1; off >>= 1) {
      float ov = __shfl_xor(v, off);
      int os = __shfl_xor(src, off);
      if (ov < v || (ov == v && os < src)) { v = ov; src = os; }
    }
    int myidx = 0;
    if (lane == src) {
      myidx = __float_as_int(smem[mbase + 2 * hp + 1]);
      hp++;
    }
    int widx = __shfl(myidx, src);
    float w = expf(-sqrtf(fmaxf(v, 1e-12f)));  // cdist = sqrt(max(d2, eps))
    if (lane < C_LAB) acc = fmaf(labels[(size_t)widx * C_LAB + lane], w, acc);
  }
  if (lane < C_LAB) out[(size_t)(qbase + wave) * C_LAB + lane] = logf(acc);
}

// ---------------------------------------------------------------------------
extern "C" void kernel_launch(void* const* d_in, const int* in_sizes, int n_in,
                              void* d_out, int out_size, void* d_ws,
                              size_t ws_size, hipStream_t stream) {
  const float* x         = (const float*)d_in[0];  // [2048, 3072]
  const float* raw       = (const float*)d_in[1];  // [50000, 3072]
  const float* projector = (const float*)d_in[2];  // [3072, 30]
  const float* labels    = (const float*)d_in[3];  // [50000, 10]
  float* out = (float*)d_out;                      // [2048, 10]

  float* ws = (float*)d_ws;
  size_t off = 0;
  float* projP   = ws + off; off += (size_t)D_DIM * P_PAD;
  float* psum    = ws + off; off += 32;
  float* meanv   = ws + off; off += 32;
  float* rawsum  = ws + off; off += N_REF;
  float* rawsq   = ws + off; off += N_REF;
  float* rawproj = ws + off; off += (size_t)N_REF * P_PAD;
  float* data    = ws + off; off += (size_t)N_PAD * P_PAD;
  float* dd      = ws + off; off += N_PAD;
  float* xsum    = ws + off; off += B_Q;
  float* xsq     = ws + off; off += B_Q;
  float* xproj   = ws + off; off += (size_t)B_Q * P_PAD;
  float* px      = ws + off; off += (size_t)B_Q * P_PAD;
  float* qq      = ws + off; off += B_Q;

  k_pad_projector<<<(D_DIM * P_PAD + 255) / 256, 256, 0, stream>>>(projector,
                                                                   projP);
  k_psum<<<1, 32, 0, stream>>>(projP, psum);
  k_rowpass<<<N_REF / 16, 32, 0, stream>>>(raw, projP, rawproj, rawsum, rawsq);
  k_mean<<<1, 1024, 0, stream>>>(rawsum, meanv);
  k_fixup_data<<<(N_PAD + 255) / 256, 256, 0, stream>>>(rawproj, rawsum, rawsq,
                                                        psum, meanv, data, dd);
  k_rowpass<<<B_Q / 16, 32, 0, stream>>>(x, projP, xproj, xsum, xsq);
  k_fixup_x<<<(B_Q + 255) / 256, 256, 0, stream>>>(xproj, xsum, xsq, psum, px,
                                                   qq);
  k_knn<<<B_Q / 16, 512, 0, stream>>>(px, qq, data, dd, labels, out);

  (void)in_sizes; (void)n_in; (void)out_size; (void)ws_size;
}